// IILoss_1906965479790
// MI455X (gfx1250) — compile-verified
//
#include <hip/hip_runtime.h>
#include <hip/hip_bf16.h>

typedef __attribute__((ext_vector_type(2))) float v2f;
typedef __attribute__((ext_vector_type(8))) float v8f;

#define N_PTS 65536
#define DIM   512
#define NCLS  1000
#define CP    1008          // NCLS padded to multiple of 16
#define NTILE (CP / 16)     // 63
#define DELTA_F 1e9f

// ---- order-preserving float <-> uint mapping (for atomicMin on floats) ----
__device__ __forceinline__ unsigned orderedEnc(float f) {
    unsigned u = __float_as_uint(f);
    return (u & 0x80000000u) ? ~u : (u | 0x80000000u);
}
__device__ __forceinline__ float orderedDec(unsigned e) {
    unsigned u = (e & 0x80000000u) ? (e ^ 0x80000000u) : ~e;
    return __uint_as_float(u);
}

// ws layout (floats): sums/means [CP*DIM] | counts [CP] | sqv [CP] | minU [1] | partials [N_PTS]
#define MINU_IDX ((size_t)CP * DIM + 2 * CP)

// ---- 0) init: zero sums/counts/sqv, seed minU with encoded +inf ----
__global__ void init_kernel(float* __restrict__ ws) {
    size_t total = MINU_IDX + 1;
    for (size_t i = (size_t)blockIdx.x * blockDim.x + threadIdx.x; i < total;
         i += (size_t)gridDim.x * blockDim.x) {
        if (i == MINU_IDX) ((unsigned*)ws)[i] = orderedEnc(__builtin_inff());
        else               ws[i] = 0.0f;
    }
}

// ---- 1) segmented sums + counts via L2-resident fp32 atomics ----
__global__ __launch_bounds__(128)
void accum_kernel(const float* __restrict__ emb, const int* __restrict__ labels,
                  float* __restrict__ sums, float* __restrict__ counts) {
    const int row = blockIdx.x;
    const int lab = labels[row];
    const float4 e = ((const float4*)(emb + (size_t)row * DIM))[threadIdx.x];
    float* s = sums + (size_t)lab * DIM + threadIdx.x * 4;
    atomicAdd(s + 0, e.x);
    atomicAdd(s + 1, e.y);
    atomicAdd(s + 2, e.z);
    atomicAdd(s + 3, e.w);
    if (threadIdx.x == 0) atomicAdd(counts + lab, 1.0f);
}

// ---- 2) means = sums / max(counts,1), in place (pad rows stay 0) ----
__global__ __launch_bounds__(256)
void means_kernel(float* __restrict__ sums, const float* __restrict__ counts) {
    const int idx = blockIdx.x * 256 + threadIdx.x;    // < CP*DIM
    const int c = idx / DIM;
    sums[idx] = sums[idx] / fmaxf(counts[c], 1.0f);
}

// ---- 3) per-class squared norm, with validity folded in:
//         sqv[c] = valid ? ||mean_c||^2 : +inf  (branchless downstream mask) ----
__global__ __launch_bounds__(128)
void sqnorm_kernel(const float* __restrict__ means, const float* __restrict__ counts,
                   float* __restrict__ sqv) {
    __shared__ float red[128];
    const int c = blockIdx.x;
    const float4 m = ((const float4*)(means + (size_t)c * DIM))[threadIdx.x];
    red[threadIdx.x] = m.x * m.x + m.y * m.y + m.z * m.z + m.w * m.w;
    __syncthreads();
    for (int o = 64; o > 0; o >>= 1) {
        if (threadIdx.x < o) red[threadIdx.x] += red[threadIdx.x + o];
        __syncthreads();
    }
    if (threadIdx.x == 0) {
        const bool valid = (c < NCLS) && (counts[c] > 0.0f);
        sqv[c] = valid ? red[0] : __builtin_inff();
    }
}

// ---- 4) intra spread: per-row squared distance to class mean ----
__global__ __launch_bounds__(128)
void intra_kernel(const float* __restrict__ emb, const int* __restrict__ labels,
                  const float* __restrict__ means, float* __restrict__ partials) {
    __shared__ float red[128];
    const int row = blockIdx.x;
    const int lab = labels[row];
    const float4 e  = ((const float4*)(emb   + (size_t)row * DIM))[threadIdx.x];
    const float4 mu = ((const float4*)(means + (size_t)lab * DIM))[threadIdx.x];
    const float dx = e.x - mu.x, dy = e.y - mu.y, dz = e.z - mu.z, dw = e.w - mu.w;
    red[threadIdx.x] = dx * dx + dy * dy + dz * dz + dw * dw;
    __syncthreads();
    for (int o = 64; o > 0; o >>= 1) {
        if (threadIdx.x < o) red[threadIdx.x] += red[threadIdx.x + o];
        __syncthreads();
    }
    if (threadIdx.x == 0) partials[row] = red[0];
}

// ---- 5) Gram tiles via fp32 WMMA, branchless masked pairwise-distance min ----
// A 16x4 fp32 layout: lane = 16*(k-pair half) + M, v2f = {K, K+1}
// B 4x16 fp32 layout: transpose-symmetric -> same per-lane pattern on B's rows
// D 16x16 fp32 layout: VGPR v -> row v + 8*(lane>>4), col = lane & 15
__global__ __launch_bounds__(32)
void gram_min_kernel(const float* __restrict__ means, const float* __restrict__ sqv,
                     unsigned* __restrict__ minU) {
    const int ti = blockIdx.x / NTILE;
    const int tj = blockIdx.x % NTILE;
    if (ti > tj) return;               // symmetric: uniform wave-level skip, EXEC stays all-1s

    const int lane = threadIdx.x;      // 0..31 (wave32)
    const int m = lane & 15;
    const int h = lane >> 4;

    const float* Arow = means + (size_t)(ti * 16 + m) * DIM + 2 * h;
    const float* Brow = means + (size_t)(tj * 16 + m) * DIM + 2 * h;

    v8f acc = {0.f, 0.f, 0.f, 0.f, 0.f, 0.f, 0.f, 0.f};
    #pragma unroll 8
    for (int k = 0; k < DIM; k += 4) {
        v2f a = *(const v2f*)(Arow + k);
        v2f b = *(const v2f*)(Brow + k);
        acc = __builtin_amdgcn_wmma_f32_16x16x4_f32(
            /*neg_a=*/false, a, /*neg_b=*/false, b,
            /*c_mod=*/(short)0, acc, /*reuse_a=*/false, /*reuse_b=*/false);
    }

    const int gj = tj * 16 + m;
    const float sqj = sqv[gj];         // +inf if class invalid/empty/padded

    float lmin = __builtin_inff();
    #pragma unroll
    for (int v = 0; v < 8; ++v) {
        const int gi = ti * 16 + v + 8 * h;
        float d2 = sqv[gi] + sqj - 2.0f * acc[v];      // inf-propagating mask
        d2 = (gi == gj) ? __builtin_inff() : d2;        // branchless diagonal mask
        lmin = fminf(lmin, d2);
    }
    #pragma unroll
    for (int off = 16; off > 0; off >>= 1)
        lmin = fminf(lmin, __shfl_xor(lmin, off, 32));
    if (lane == 0) atomicMin(minU, orderedEnc(lmin));
}

// ---- 6) final combine: deterministic sum of partials + min decode ----
__global__ __launch_bounds__(256)
void final_kernel(const float* __restrict__ partials, const unsigned* __restrict__ minU,
                  float* __restrict__ out) {
    __shared__ float red[256];
    float s = 0.0f;
    for (int i = threadIdx.x; i < N_PTS; i += 256) s += partials[i];
    red[threadIdx.x] = s;
    __syncthreads();
    for (int o = 128; o > 0; o >>= 1) {
        if (threadIdx.x < o) red[threadIdx.x] += red[threadIdx.x + o];
        __syncthreads();
    }
    if (threadIdx.x == 0) {
        const float inter = orderedDec(*minU);
        out[0] = red[0] / (float)N_PTS - fminf(DELTA_F, inter);
    }
}

extern "C" void kernel_launch(void* const* d_in, const int* in_sizes, int n_in,
                              void* d_out, int out_size, void* d_ws, size_t ws_size,
                              hipStream_t stream) {
    const float* emb    = (const float*)d_in[0];
    const int*   labels = (const int*)d_in[1];
    (void)in_sizes; (void)n_in; (void)out_size; (void)ws_size;

    float*    ws       = (float*)d_ws;
    float*    sums     = ws;                               // [CP*DIM] -> becomes means
    float*    counts   = sums + (size_t)CP * DIM;          // [CP]
    float*    sqv      = counts + CP;                      // [CP]
    unsigned* minU     = (unsigned*)(sqv + CP);            // [1]
    float*    partials = (float*)(minU + 1);               // [N_PTS]
    float*    out      = (float*)d_out;

    init_kernel<<<512, 256, 0, stream>>>(ws);
    accum_kernel<<<N_PTS, 128, 0, stream>>>(emb, labels, sums, counts);
    means_kernel<<<(CP * DIM) / 256, 256, 0, stream>>>(sums, counts);
    sqnorm_kernel<<<CP, 128, 0, stream>>>(sums, counts, sqv);
    intra_kernel<<<N_PTS, 128, 0, stream>>>(emb, labels, sums, partials);
    gram_min_kernel<<<NTILE * NTILE, 32, 0, stream>>>(sums, sqv, minU);
    final_kernel<<<1, 256, 0, stream>>>(partials, minU, out);
}